// Centroids_25271587570291
// MI455X (gfx1250) — compile-verified
//
#include <hip/hip_runtime.h>

#define N_PTS 131072
#define DIMS  64
#define KCENT 1024
#define DECAY 0.99f
#define OMD   0.01f   /* 1 - DECAY */
#define EPSV  1e-5f

typedef float v2f __attribute__((ext_vector_type(2)));
typedef float v4f __attribute__((ext_vector_type(4)));
typedef float v8f __attribute__((ext_vector_type(8)));
typedef int   v4i __attribute__((ext_vector_type(4)));

#if defined(__AMDGCN__) && \
    __has_builtin(__builtin_amdgcn_global_load_async_to_lds_b128) && \
    __has_builtin(__builtin_amdgcn_s_wait_asynccnt)
#define USE_ASYNC_LDS 1
typedef __attribute__((address_space(1))) v4i gv4i;   // global (__device__) v4i
typedef __attribute__((address_space(3))) v4i lv4i;   // LDS (__shared__) v4i
#else
#define USE_ASYNC_LDS 0
#endif

// ---------------------------------------------------------------------------
// Kernel A: c_sq[k] = sum_d C[d,k]^2 ; zero counts, csum, loss slot.
// grid = 256 x 256  (65536 threads == D*K)
// ---------------------------------------------------------------------------
__global__ __launch_bounds__(256) void prep_kernel(
    const float* __restrict__ cent, float* __restrict__ csq,
    float* __restrict__ counts, float* __restrict__ csum,
    float* __restrict__ loss_slot)
{
    int g = blockIdx.x * 256 + threadIdx.x;      // 0 .. 65535
    csum[g] = 0.0f;
    if (g < KCENT) {
        float s = 0.0f;
        #pragma unroll
        for (int d = 0; d < DIMS; ++d) {
            float v = cent[d * KCENT + g];
            s = fmaf(v, v, s);
        }
        csq[g]    = s;
        counts[g] = 0.0f;
    }
    if (g == 0) *loss_slot = 0.0f;
}

// ---------------------------------------------------------------------------
// Kernel B: fused distance GEMM (fp32 WMMA 16x16x4) + argmin + segment sums.
// 256 threads = 8 waves; block handles 128 rows x all 1024 centroids.
// Centroid chunks (64x128) double-buffered in LDS via async global->LDS copy.
// ---------------------------------------------------------------------------
__global__ __launch_bounds__(256) void assign_kernel(
    const float* __restrict__ x, const float* __restrict__ cent,
    const float* __restrict__ csq, float* __restrict__ counts,
    float* __restrict__ csum, int* __restrict__ ind)
{
    __shared__ float ldsC[2][64 * 136];  // 64 rows (D) x 128 cols, pad->136, x2 buf
    __shared__ float redV[128 * 16];
    __shared__ int   redI[128 * 16];

    const int tid     = threadIdx.x;
    const int wave    = tid >> 5;
    const int lane    = tid & 31;
    const int lane_lo = lane & 15;
    const int half    = lane >> 4;
    const int base_m  = blockIdx.x * 128 + wave * 16;

    // --- A tile: 16 rows x 64 cols of x, resident in registers (WMMA layout).
    // lane<16: row=lane, K = 4k+{0,1}; lane>=16: row=lane-16, K = 4k+{2,3}.
    v2f a[16];
    const float* xrow = x + (size_t)(base_m + lane_lo) * DIMS + 2 * half;
    #pragma unroll
    for (int kc = 0; kc < 16; ++kc)
        a[kc] = *(const v2f*)(xrow + kc * 4);   // 8B-aligned global_load_b64

    // --- per-row ||x||^2 from the register tile
    float part = 0.0f;
    #pragma unroll
    for (int kc = 0; kc < 16; ++kc)
        part = fmaf(a[kc].x, a[kc].x, fmaf(a[kc].y, a[kc].y, part));
    float xsq = part + __shfl_xor(part, 16, 32);     // combine the two halves
    float vsq[8];
    #pragma unroll
    for (int r = 0; r < 8; ++r)                       // C/D row layout broadcast
        vsq[r] = __shfl(xsq, r + (half << 3), 32);

    float bestv[8];
    int   besti[8];
    #pragma unroll
    for (int r = 0; r < 8; ++r) { bestv[r] = 3.0e38f; besti[r] = 0; }

    // --- chunk loader: 64 x 128 floats = 2048 float4, 8 per thread ----------
    auto load_chunk = [&](int kb, int buf) {
        #pragma unroll
        for (int it = 0; it < 8; ++it) {
            int idx = it * 256 + tid;                 // 0..2047 float4 slots
            int d = idx >> 5, c4 = idx & 31;
            const float* g = cent + d * KCENT + kb * 128 + c4 * 4;
            float*       l = &ldsC[buf][d * 136 + c4 * 4];
#if USE_ASYNC_LDS
            __builtin_amdgcn_global_load_async_to_lds_b128(
                (gv4i*)(void*)g, (lv4i*)(void*)l, 0, 0);
#else
            *(v4f*)l = *(const v4f*)g;
#endif
        }
    };
    auto wait_copies = [&]() {
#if USE_ASYNC_LDS
        __builtin_amdgcn_s_wait_asynccnt(0);
#endif
    };

    load_chunk(0, 0);                                 // prime the pipeline
    wait_copies();
    __syncthreads();

    for (int kb = 0; kb < 8; ++kb) {                  // 8 chunks of 128 centroids
        const int cur = kb & 1;
        if (kb < 7) load_chunk(kb + 1, cur ^ 1);      // prefetch next chunk

        #pragma unroll 1
        for (int nt = 0; nt < 8; ++nt) {              // 8 N-tiles of 16 cols
            const int n0 = nt * 16;
            // prefetch all B operands for this tile into registers first,
            // so the 16 WMMAs below issue back-to-back without DS waits
            const float* lc = &ldsC[cur][(2 * half) * 136 + n0 + lane_lo];
            v2f b[16];
            #pragma unroll
            for (int kc = 0; kc < 16; ++kc) {
                b[kc].x = lc[kc * 4 * 136];           // K row 4kc + 2h
                b[kc].y = lc[kc * 4 * 136 + 136];     // K row 4kc + 2h + 1
            }
            v8f acc = {0.f,0.f,0.f,0.f,0.f,0.f,0.f,0.f};
            #pragma unroll
            for (int kc = 0; kc < 16; ++kc)
                acc = __builtin_amdgcn_wmma_f32_16x16x4_f32(
                        false, a[kc], false, b[kc], (short)0, acc, false, false);

            const int   col = kb * 128 + n0 + lane_lo;
            const float vc  = csq[col];
            #pragma unroll
            for (int r = 0; r < 8; ++r) {
                float dist = fmaf(-2.0f, acc[r], vsq[r] + vc);
                if (dist < bestv[r] || (dist == bestv[r] && col < besti[r])) {
                    bestv[r] = dist; besti[r] = col;
                }
            }
        }

        wait_copies();                                // our prefetch landed
        __syncthreads();                              // everyone's landed
    }

    #pragma unroll
    for (int r = 0; r < 8; ++r) {
        int row_local = wave * 16 + r + 8 * half;     // C/D layout: M = r + 8*half
        redV[row_local * 16 + lane_lo] = bestv[r];
        redI[row_local * 16 + lane_lo] = besti[r];
    }
    __syncthreads();

    if (tid < 128) {                                   // final 16-way argmin / row
        float bv = redV[tid * 16];
        int   bi = redI[tid * 16];
        #pragma unroll
        for (int l = 1; l < 16; ++l) {
            float v  = redV[tid * 16 + l];
            int   i2 = redI[tid * 16 + l];
            if (v < bv || (v == bv && i2 < bi)) { bv = v; bi = i2; }
        }
        const int row = blockIdx.x * 128 + tid;
        ind[row] = bi;
        atomicAdd(&counts[bi], 1.0f);                  // segment_sum(ones)
        const float* xr = x + (size_t)row * DIMS;
        #pragma unroll
        for (int d = 0; d < DIMS; ++d)                 // segment_sum(x) -> [D,K]
            atomicAdd(&csum[d * KCENT + bi], xr[d]);
    }
}

// ---------------------------------------------------------------------------
// Kernel C: EMA update + Laplace smoothing + write transposed codebook [K,D].
// single block of 1024 threads (one per centroid)
// ---------------------------------------------------------------------------
__global__ __launch_bounds__(1024) void finalize_kernel(
    const float* __restrict__ cluster_size, const float* __restrict__ cavg,
    const float* __restrict__ counts, const float* __restrict__ csum,
    float* __restrict__ cnewT)
{
    __shared__ float sred[1024];
    const int k = threadIdx.x;
    float csn = DECAY * cluster_size[k] + OMD * counts[k];
    sred[k] = csn;
    __syncthreads();
    for (int s = 512; s > 0; s >>= 1) {
        if (k < s) sred[k] += sred[k + s];
        __syncthreads();
    }
    const float n   = sred[0];
    const float cs  = (csn + EPSV) / (n + (float)KCENT * EPSV) * n;
    const float inv = 1.0f / cs;
    #pragma unroll
    for (int d = 0; d < DIMS; ++d) {
        float av = DECAY * cavg[d * KCENT + k] + OMD * csum[d * KCENT + k];
        cnewT[k * DIMS + d] = av * inv;   // transposed for coalesced gather
    }
}

// ---------------------------------------------------------------------------
// Kernel D: x_quantized gather (== straight-through output) + MSE loss.
// ---------------------------------------------------------------------------
__global__ __launch_bounds__(256) void gather_kernel(
    const float* __restrict__ x, const int* __restrict__ ind,
    const float* __restrict__ cnewT, float* __restrict__ out,
    float* __restrict__ loss)
{
    __shared__ float sred[256];
    const long long total  = (long long)N_PTS * DIMS;
    const int       tid    = threadIdx.x;
    long long       g      = (long long)blockIdx.x * 256 + tid;
    const long long stride = (long long)gridDim.x * 256;
    float part = 0.0f;
    for (; g < total; g += stride) {
        const int i = (int)(g >> 6);
        const int d = (int)(g & 63);
        const int c = ind[i];
        const float q  = cnewT[c * DIMS + d];
        const float xv = x[g];
        out[g] = q;                        // x + sg(x_q - x) == x_q numerically
        const float diff = xv - q;
        part = fmaf(diff, diff, part);
    }
    sred[tid] = part;
    __syncthreads();
    for (int s = 128; s > 0; s >>= 1) {
        if (tid < s) sred[tid] += sred[tid + s];
        __syncthreads();
    }
    if (tid == 0)
        atomicAdd(loss, sred[0] * (1.0f / ((float)N_PTS * (float)DIMS)));
}

// ---------------------------------------------------------------------------
extern "C" void kernel_launch(void* const* d_in, const int* in_sizes, int n_in,
                              void* d_out, int out_size, void* d_ws, size_t ws_size,
                              hipStream_t stream)
{
    const float* x            = (const float*)d_in[0];   // [N,D]
    const float* cent         = (const float*)d_in[1];   // [D,K]
    const float* cluster_size = (const float*)d_in[2];   // [K]
    const float* cavg         = (const float*)d_in[3];   // [D,K]

    float* out  = (float*)d_out;                         // [N*D] + 1 (loss)
    float* loss = out + (size_t)N_PTS * DIMS;

    // workspace layout (floats): csq[1024] | counts[1024] | csum[65536]
    //                            | ind[131072 ints] | cnewT[65536]
    float* ws     = (float*)d_ws;
    float* csq    = ws;
    float* counts = ws + 1024;
    float* csum   = ws + 2048;
    int*   ind    = (int*)(ws + 2048 + (size_t)DIMS * KCENT);
    float* cnewT  = ws + 2048 + (size_t)DIMS * KCENT + N_PTS;

    prep_kernel    <<<256, 256, 0, stream>>>(cent, csq, counts, csum, loss);
    assign_kernel  <<<N_PTS / 128, 256, 0, stream>>>(x, cent, csq, counts, csum, ind);
    finalize_kernel<<<1, 1024, 0, stream>>>(cluster_size, cavg, counts, csum, cnewT);
    gather_kernel  <<<4096, 256, 0, stream>>>(x, ind, cnewT, out, loss);
}